// Filterbank_18167711662157
// MI455X (gfx1250) — compile-verified
//
#include <hip/hip_runtime.h>

#define NB         32
#define TLEN       16000
#define KT         9
#define NBLK       (TLEN / 16)
#define BANDSTRIDE 512   // per-band floats in ws: 384 (U 16x24) + 128 (V 16x8)

typedef __attribute__((ext_vector_type(2))) float v2f;
typedef __attribute__((ext_vector_type(4))) float v4f;
typedef __attribute__((ext_vector_type(8))) float v8f;

// ---------------------------------------------------------------------------
// Setup (f64): per band, build block state-space operators.
//   y[t] = xf[t] - sum_{m=1..8} a[m] y[t-m],  xf = FIR(b) of raw input.
//   W (16x16): block-xf -> block-y impulse operator of 1/A(z)
//   V (16x8) : incoming 8-state -> block-y
//   U = W @ B_fir (16x24): RAW window z[s0-8 .. s0+15] -> block-y
// The FIR is folded into U so the whole per-block update is pure WMMA.
// ---------------------------------------------------------------------------
__global__ void fb_setup(const double* __restrict__ a, const double* __restrict__ b,
                         float* __restrict__ ws)
{
    int band = threadIdx.x;
    if (band >= NB) return;

    double an[KT], bn[KT];
    double a0 = a[band * KT];
    for (int k = 0; k < KT; ++k) {
        an[k] = a[band * KT + k] / a0;
        bn[k] = b[band * KT + k] / a0;
    }

    double W[16][16], V[16][8];
    for (int t = 0; t < 16; ++t) {
        for (int j = 0; j < 16; ++j) W[t][j] = (j == t) ? 1.0 : 0.0;
        for (int c = 0; c < 8; ++c) {
            int m = t + c + 1;                 // direct reach into incoming state
            V[t][c] = (m <= 8) ? -an[m] : 0.0;
        }
        for (int m = 1; m <= 8; ++m) {
            if (t - m >= 0) {
                for (int j = 0; j < 16; ++j) W[t][j] -= an[m] * W[t - m][j];
                for (int c = 0; c < 8; ++c)  V[t][c] -= an[m] * V[t - m][c];
            }
        }
    }

    // U[t][j] = sum_i W[t][i] * B_fir[i][j], B_fir[i][j] = bn[i+8-j] (banded)
    float* base = ws + band * BANDSTRIDE;
    for (int t = 0; t < 16; ++t)
        for (int j = 0; j < 24; ++j) {
            double s = 0.0;
            for (int i = 0; i < 16; ++i) {
                int k = i + 8 - j;
                if (k >= 0 && k < KT) s += W[t][i] * bn[k];
            }
            base[t * 24 + j] = (float)s;
        }
    for (int t = 0; t < 16; ++t)
        for (int c = 0; c < 8; ++c) base[384 + t * 8 + c] = (float)V[t][c];
}

// ---------------------------------------------------------------------------
// One IIR pass. One wave per band; the 16 batches fill the WMMA N dimension.
// Per 16-sample block:  Y = U @ Z_raw (6 WMMAs, K=24)  +  V @ S (2 WMMAs).
// REV=0: src is signal (16,T) -> dst.  REV=1: src==dst (16,NB,T), processed
// time-reversed in place (next window is loaded BEFORE current stores).
// ---------------------------------------------------------------------------
template <int REV>
__global__ __launch_bounds__(32)
void fb_pass(const float* src, float* dst, const float* __restrict__ coef)
{
    const int band = blockIdx.x;
    const int lane = threadIdx.x;
    const int low  = lane & 15;   // A: row M, B: col N (batch), C: col N
    const int hi   = lane >> 4;   // selects K / M sub-half of fragments

    const float* cb = coef + band * BANDSTRIDE;

    // A fragments: U (16x24 -> 6 K-chunks), V (16x8 -> 2 K-chunks).
    v2f aU[6];
#pragma unroll
    for (int c = 0; c < 6; ++c) {
        int k = 4 * c + 2 * hi;
        aU[c].x = cb[low * 24 + k];
        aU[c].y = cb[low * 24 + k + 1];
    }
    v2f aV[2];
#pragma unroll
    for (int c = 0; c < 2; ++c) {
        int k = 4 * c + 2 * hi;
        aV[c].x = cb[384 + low * 8 + k];
        aV[c].y = cb[384 + low * 8 + k + 1];
    }

    // Logical sequence z[s]: REV=0 -> sp[s]; REV=1 -> sp[-s] (sp at row end).
    const float* sp = REV ? src + ((long)low * NB + band) * (long)TLEN + (TLEN - 1)
                          : src + (long)low * TLEN;
    float* dp = dst + ((long)low * NB + band) * (long)TLEN;

    auto LD = [&](int e) -> float { return REV ? sp[-(long)e] : sp[(long)e]; };

    // Incoming state B-fragment values: S[r][n] = y[s0-(r+1)]; this lane holds
    // r = 2hi, 2hi+1 (chunk 0) and 4+2hi, 5+2hi (chunk 1).
    float st0 = 0.f, st1 = 0.f, st2 = 0.f, st3 = 0.f;

    // Raw-input B fragments: wv[2c+d] = z[s0 - 8 + 4c + 2hi + d].
    // Only block 0 can reach s < 0 -> guards live only in this prologue.
    float wv[12];
#pragma unroll
    for (int c = 0; c < 6; ++c)
#pragma unroll
        for (int d = 0; d < 2; ++d) {
            int s = -8 + 4 * c + 2 * hi + d;
            wv[2 * c + d] = (s >= 0) ? LD(s) : 0.f;
        }

    int s0 = 0;
#pragma unroll 1
    for (int blk = 0; blk < NBLK; ++blk) {
        // Y = U @ Z_raw  (state-independent, off the carried critical path)
        v8f acc = {0.f, 0.f, 0.f, 0.f, 0.f, 0.f, 0.f, 0.f};
#pragma unroll
        for (int c = 0; c < 6; ++c) {
            v2f bz; bz.x = wv[2 * c]; bz.y = wv[2 * c + 1];
            acc = __builtin_amdgcn_wmma_f32_16x16x4_f32(false, aU[c], false, bz,
                                                        (short)0, acc, false, false);
        }
        // Y += V @ S  (carried dependency: 2 WMMAs per 16 samples)
        {
            v2f bs; bs.x = st0; bs.y = st1;
            acc = __builtin_amdgcn_wmma_f32_16x16x4_f32(false, aV[0], false, bs,
                                                        (short)0, acc, false, false);
            bs.x = st2; bs.y = st3;
            acc = __builtin_amdgcn_wmma_f32_16x16x4_f32(false, aV[1], false, bs,
                                                        (short)0, acc, false, false);
        }

        // Preload the NEXT window ([s0+8, s0+31], always in range) BEFORE the
        // stores below: the reverse pass runs in place over these addresses.
        float wn[12];
        if (blk + 1 < NBLK) {
            const int e0 = s0 + 8 + 2 * hi;
#pragma unroll
            for (int c = 0; c < 6; ++c)
#pragma unroll
                for (int d = 0; d < 2; ++d)
                    wn[2 * c + d] = LD(e0 + 4 * c + d);
            __builtin_prefetch(REV ? (const void*)(sp - (long)(e0 + 40))
                                   : (const void*)(sp + (long)(e0 + 40)), 0, 1);
        } else {
#pragma unroll
            for (int i = 0; i < 12; ++i) wn[i] = 0.f;
        }

        // Store Y: lane holds t = r + 8*hi (r=0..7) for batch `low`; 8
        // consecutive samples either direction -> two aligned b128 stores.
        if (REV) {
            int p3 = TLEN - 4 - s0 - 8 * hi;           // pos of t = r+8hi, r=3
            v4f lo4 = {acc[3], acc[2], acc[1], acc[0]};
            v4f hi4 = {acc[7], acc[6], acc[5], acc[4]};
            *reinterpret_cast<v4f*>(dp + p3)     = lo4;
            *reinterpret_cast<v4f*>(dp + p3 - 4) = hi4;
        } else {
            int p0 = s0 + 8 * hi;
            v4f lo4 = {acc[0], acc[1], acc[2], acc[3]};
            v4f hi4 = {acc[4], acc[5], acc[6], acc[7]};
            *reinterpret_cast<v4f*>(dp + p0)     = lo4;
            *reinterpret_cast<v4f*>(dp + p0 + 4) = hi4;
        }

        // State handoff: S'[r][n] = Y[15-r][n]; Y[8+j][n] lives in lane 16+n,
        // C-VGPR j. Broadcast all 8 rows, each lane keeps the 4 it owns.
        {
            float g[8];
#pragma unroll
            for (int j = 0; j < 8; ++j) g[j] = __shfl(acc[j], 16 + low, 32);
            st0 = hi ? g[5] : g[7];   // r = 2hi
            st1 = hi ? g[4] : g[6];   // r = 2hi+1
            st2 = hi ? g[1] : g[3];   // r = 4+2hi
            st3 = hi ? g[0] : g[2];   // r = 5+2hi
        }

#pragma unroll
        for (int i = 0; i < 12; ++i) wv[i] = wn[i];
        s0 += 16;
    }
}

// ---------------------------------------------------------------------------
// Launch: setup -> forward pass (signal -> out) -> reversed pass (out in place)
// ---------------------------------------------------------------------------
extern "C" void kernel_launch(void* const* d_in, const int* in_sizes, int n_in,
                              void* d_out, int out_size, void* d_ws, size_t ws_size,
                              hipStream_t stream)
{
    const float*  signal = (const float*)d_in[0];
    const double* a      = (const double*)d_in[1];
    const double* b      = (const double*)d_in[2];
    float*        out    = (float*)d_out;
    float*        ws     = (float*)d_ws;   // 32*512 floats = 64 KB

    fb_setup<<<1, 32, 0, stream>>>(a, b, ws);
    fb_pass<0><<<NB, 32, 0, stream>>>(signal, out, ws);  // forward lfilter
    fb_pass<1><<<NB, 32, 0, stream>>>(out,    out, ws);  // reversed, in place
}